// RobustGNN_77326591197817
// MI455X (gfx1250) — compile-verified
//
#include <hip/hip_runtime.h>
#include <hip/hip_bf16.h>

typedef __attribute__((ext_vector_type(2))) float v2f;
typedef __attribute__((ext_vector_type(8))) float v8f;

#define HIDF 128
#define BN_EPS 1e-5f

// ---------------------------------------------------------------------------
// Generic zero
// ---------------------------------------------------------------------------
__global__ void k_zero(float* __restrict__ p, int n) {
    int i = blockIdx.x * blockDim.x + threadIdx.x;
    if (i < n) p[i] = 0.0f;
}

// ---------------------------------------------------------------------------
// Degree: deg = 1 (self loop) + count of incoming edges; dinv = rsqrt(deg)
// ---------------------------------------------------------------------------
__global__ void k_deg_init(float* __restrict__ deg, int n) {
    int i = blockIdx.x * blockDim.x + threadIdx.x;
    if (i < n) deg[i] = 1.0f;   // self loop
}

__global__ void k_deg_edges(const int* __restrict__ dst, float* __restrict__ deg, int e) {
    int i = blockIdx.x * blockDim.x + threadIdx.x;
    if (i < e) unsafeAtomicAdd(&deg[dst[i]], 1.0f);
}

__global__ void k_dinv(const float* __restrict__ deg, float* __restrict__ dinv, int n) {
    int i = blockIdx.x * blockDim.x + threadIdx.x;
    if (i < n) {
        float d = deg[i];
        dinv[i] = (d > 0.0f) ? rsqrtf(d) : 0.0f;
    }
}

// ---------------------------------------------------------------------------
// GEMM: XW = X[n_rows,128] @ W[128,128] via V_WMMA_F32_16X16X4_F32.
// Block = 256 threads = 8 waves; block covers 32 rows (two 16-row M tiles),
// wave w owns output columns [16w, 16w+16). Each wave keeps two accumulators
// sharing one B fragment per k-step (2 WMMAs per B load pair). A staged in
// LDS (16 KB), W read from global (64 KB, L0/L2 resident).
// REQUIRES n_rows % 32 == 0 (true for N=100000) -> no bounds checks, EXEC
// stays all-ones as WMMA requires.
// ---------------------------------------------------------------------------
__global__ __launch_bounds__(256) void k_gemm_wmma(const float* __restrict__ X,
                                                   const float* __restrict__ W,
                                                   float* __restrict__ XW,
                                                   int sanitize) {
    __shared__ float sA[32 * HIDF];     // 16 KB
    const int tid  = threadIdx.x;
    const int wave = tid >> 5;          // 0..7  -> N tile
    const int lane = tid & 31;
    const int l16  = lane & 15;
    const int hs   = lane >> 4;         // half-wave select
    const int row0 = blockIdx.x * 32;

    // Stage 32x128 A tile, coalesced, with NaN scrub (nan_to_num on x)
    for (int i = tid; i < 32 * HIDF; i += 256) {
        float v = X[(size_t)row0 * HIDF + i];
        if (sanitize && (v != v)) v = 0.0f;
        sA[i] = v;
    }
    __syncthreads();

    const int nbase = wave * 16;
    v8f acc0 = {0.f, 0.f, 0.f, 0.f, 0.f, 0.f, 0.f, 0.f};
    v8f acc1 = {0.f, 0.f, 0.f, 0.f, 0.f, 0.f, 0.f, 0.f};

    #pragma unroll 4
    for (int k0 = 0; k0 < HIDF; k0 += 4) {
        const int ka = k0 + 2 * hs;     // lanes 0-15: K=k0,k0+1 ; lanes 16-31: K=k0+2,k0+3
        v2f a0, a1, b;
        b.x  = W[(size_t)ka * HIDF + nbase + l16];
        b.y  = W[(size_t)(ka + 1) * HIDF + nbase + l16];
        a0.x = sA[l16 * HIDF + ka];
        a0.y = sA[l16 * HIDF + ka + 1];
        a1.x = sA[(16 + l16) * HIDF + ka];
        a1.y = sA[(16 + l16) * HIDF + ka + 1];
        acc0 = __builtin_amdgcn_wmma_f32_16x16x4_f32(false, a0, false, b,
                                                     (short)0, acc0, false, false);
        acc1 = __builtin_amdgcn_wmma_f32_16x16x4_f32(false, a1, false, b,
                                                     (short)0, acc1, false, false);
    }

    // D layout: VGPR r -> M = r (lanes 0-15) / r+8 (lanes 16-31), N = l16
    #pragma unroll
    for (int r = 0; r < 8; ++r) {
        int gr0 = row0 + r + 8 * hs;
        XW[(size_t)gr0 * HIDF + nbase + l16]        = acc0[r];
        XW[(size_t)(gr0 + 16) * HIDF + nbase + l16] = acc1[r];
    }
}

// ---------------------------------------------------------------------------
// agg init with self-loop term: agg[n][c] = xw[n][c] * dinv[n]^2
// ---------------------------------------------------------------------------
__global__ void k_agg_init(const float* __restrict__ xw, const float* __restrict__ dinv,
                           float* __restrict__ agg, int n) {
    int i = blockIdx.x * blockDim.x + threadIdx.x;
    int total = n * HIDF;
    if (i < total) {
        int node = i >> 7;
        float di = dinv[node];
        agg[i] = xw[i] * di * di;
    }
}

// ---------------------------------------------------------------------------
// Edge scatter: 32 lanes per edge, each lane handles 4 consecutive features.
// agg[dst] += xw[src] * dinv[src]*dinv[dst]   (hardware f32 atomics)
// ---------------------------------------------------------------------------
__global__ void k_edge_scatter(const int* __restrict__ src, const int* __restrict__ dst,
                               const float* __restrict__ dinv, const float* __restrict__ xw,
                               float* __restrict__ agg, int n_edges) {
    int t = blockIdx.x * blockDim.x + threadIdx.x;
    int e = t >> 5;
    int lane = t & 31;
    if (e >= n_edges) return;
    int s = src[e], d = dst[e];
    float w = dinv[s] * dinv[d];
    const float4 v = ((const float4*)(xw + (size_t)s * HIDF))[lane];
    float* ad = agg + (size_t)d * HIDF + 4 * lane;
    unsafeAtomicAdd(ad + 0, v.x * w);
    unsafeAtomicAdd(ad + 1, v.y * w);
    unsafeAtomicAdd(ad + 2, v.z * w);
    unsafeAtomicAdd(ad + 3, v.w * w);
}

// ---------------------------------------------------------------------------
// BN column stats: per-block partial sums over 128 rows, 1 thread per column
// ---------------------------------------------------------------------------
__global__ __launch_bounds__(128) void k_colstats(const float* __restrict__ agg,
                                                  const float* __restrict__ bias,
                                                  float* __restrict__ sums,
                                                  float* __restrict__ sumsq, int n_rows) {
    int c = threadIdx.x;
    int r0 = blockIdx.x * 128;
    int r1 = min(r0 + 128, n_rows);
    float b = bias[c];
    float s = 0.0f, s2 = 0.0f;
    for (int r = r0; r < r1; ++r) {
        float v = agg[(size_t)r * HIDF + c] + b;
        s += v;
        s2 += v * v;
    }
    unsafeAtomicAdd(&sums[c], s);
    unsafeAtomicAdd(&sumsq[c], s2);
}

// ---------------------------------------------------------------------------
// BN apply + ReLU: h = relu((agg+b - mean)*rsqrt(var+eps)*g + be)
// ---------------------------------------------------------------------------
__global__ void k_bn_relu(const float* __restrict__ agg, const float* __restrict__ bias,
                          const float* __restrict__ gamma, const float* __restrict__ beta,
                          const float* __restrict__ sums, const float* __restrict__ sumsq,
                          float* __restrict__ h, int n_rows) {
    int i = blockIdx.x * blockDim.x + threadIdx.x;
    int total = n_rows * HIDF;
    if (i >= total) return;
    int c = i & 127;
    float invN = 1.0f / (float)n_rows;
    float mean = sums[c] * invN;
    float var  = sumsq[c] * invN - mean * mean;
    float v = (agg[i] + bias[c] - mean) * rsqrtf(var + BN_EPS) * gamma[c] + beta[c];
    h[i] = fmaxf(v, 0.0f);
}

// ---------------------------------------------------------------------------
// Global mean pool accumulation: flat grid, one atomic per element
// ---------------------------------------------------------------------------
__global__ void k_pool(const float* __restrict__ h, const int* __restrict__ batch,
                       float* __restrict__ pooled, float* __restrict__ counts, int n) {
    int i = blockIdx.x * blockDim.x + threadIdx.x;
    int total = n * HIDF;
    if (i >= total) return;
    int node = i >> 7;
    int c = i & 127;
    int g = batch[node];
    unsafeAtomicAdd(&pooled[(size_t)g * HIDF + c], h[i]);
    if (c == 0) unsafeAtomicAdd(&counts[g], 1.0f);
}

// ---------------------------------------------------------------------------
// Classifier: pooled/counts -> relu(@Wc1+bc1) -> @Wc2+bc2 -> out[64,2]
// ---------------------------------------------------------------------------
__global__ __launch_bounds__(128) void k_classifier(const float* __restrict__ pooled,
                                                    const float* __restrict__ counts,
                                                    const float* __restrict__ Wc1,
                                                    const float* __restrict__ bc1,
                                                    const float* __restrict__ Wc2,
                                                    const float* __restrict__ bc2,
                                                    float* __restrict__ out) {
    __shared__ float pm[64 * HIDF];   // 32 KB
    __shared__ float z[64 * 64];      // 16 KB
    int tid = threadIdx.x;

    for (int i = tid; i < 64 * HIDF; i += 128) {
        int g = i >> 7;
        float cnt = fmaxf(counts[g], 1.0f);
        pm[i] = pooled[i] / cnt;
    }
    __syncthreads();

    for (int i = tid; i < 64 * 64; i += 128) {
        int g = i >> 6, j = i & 63;
        float acc = bc1[j];
        for (int k = 0; k < HIDF; ++k) acc += pm[g * HIDF + k] * Wc1[k * 64 + j];
        z[i] = fmaxf(acc, 0.0f);
    }
    __syncthreads();

    for (int i = tid; i < 64 * 2; i += 128) {
        int g = i >> 1, j = i & 1;
        float acc = bc2[j];
        for (int k = 0; k < 64; ++k) acc += z[g * 64 + k] * Wc2[k * 2 + j];
        if (acc != acc) acc = 0.0f;   // nan_to_num
        out[i] = acc;
    }
}

// ---------------------------------------------------------------------------
// Host launcher
// ---------------------------------------------------------------------------
extern "C" void kernel_launch(void* const* d_in, const int* in_sizes, int n_in,
                              void* d_out, int out_size, void* d_ws, size_t ws_size,
                              hipStream_t stream) {
    const float* x    = (const float*)d_in[0];
    const int*   ei   = (const int*)d_in[1];
    const int*   bat  = (const int*)d_in[2];
    const float* W1   = (const float*)d_in[3];
    const float* b1   = (const float*)d_in[4];
    const float* g1   = (const float*)d_in[5];
    const float* be1  = (const float*)d_in[6];
    const float* W2   = (const float*)d_in[7];
    const float* b2   = (const float*)d_in[8];
    const float* g2   = (const float*)d_in[9];
    const float* be2  = (const float*)d_in[10];
    const float* Wc1  = (const float*)d_in[11];
    const float* bc1  = (const float*)d_in[12];
    const float* Wc2  = (const float*)d_in[13];
    const float* bc2  = (const float*)d_in[14];
    float* out = (float*)d_out;

    const int N = in_sizes[0] / HIDF;     // 100000 (multiple of 32)
    const int E = in_sizes[1] / 2;        // 1600000
    const int* src = ei;
    const int* dst = ei + E;

    // workspace carve-up
    float* ws = (float*)d_ws;
    float* deg    = ws;                 ws += N;
    float* dinv   = ws;                 ws += N;
    float* bufA   = ws;                 ws += (size_t)N * HIDF;
    float* bufB   = ws;                 ws += (size_t)N * HIDF;
    float* bufC   = ws;                 ws += (size_t)N * HIDF;
    float* sums   = ws;                 ws += HIDF;
    float* sumsq  = ws;                 ws += HIDF;
    float* pooled = ws;                 ws += 64 * HIDF;
    float* counts = ws;                 ws += 64;

    const int T = 256;
    const int nNodeBlk  = (N + T - 1) / T;
    const int nEdgeBlk  = (E + T - 1) / T;
    const int nFeatBlk  = (N * HIDF + T - 1) / T;
    const int nScatBlk  = (int)(((size_t)E * 32 + T - 1) / T);
    const int nGemmBlk  = N / 32;         // exact: N % 32 == 0
    const int nStatBlk  = (N + 127) / 128;

    // degrees (shared by both layers)
    k_deg_init<<<nNodeBlk, T, 0, stream>>>(deg, N);
    k_deg_edges<<<nEdgeBlk, T, 0, stream>>>(dst, deg, E);
    k_dinv<<<nNodeBlk, T, 0, stream>>>(deg, dinv, N);

    // ---- layer 1 ----
    k_gemm_wmma<<<nGemmBlk, 256, 0, stream>>>(x, W1, bufA, /*sanitize=*/1);
    k_agg_init<<<nFeatBlk, T, 0, stream>>>(bufA, dinv, bufB, N);
    k_edge_scatter<<<nScatBlk, T, 0, stream>>>(src, dst, dinv, bufA, bufB, E);
    k_zero<<<1, 256, 0, stream>>>(sums, 2 * HIDF);   // sums + sumsq are adjacent
    k_colstats<<<nStatBlk, 128, 0, stream>>>(bufB, b1, sums, sumsq, N);
    k_bn_relu<<<nFeatBlk, T, 0, stream>>>(bufB, b1, g1, be1, sums, sumsq, bufC, N);

    // ---- layer 2 ----
    k_gemm_wmma<<<nGemmBlk, 256, 0, stream>>>(bufC, W2, bufA, /*sanitize=*/0);
    k_agg_init<<<nFeatBlk, T, 0, stream>>>(bufA, dinv, bufB, N);
    k_edge_scatter<<<nScatBlk, T, 0, stream>>>(src, dst, dinv, bufA, bufB, E);
    k_zero<<<1, 256, 0, stream>>>(sums, 2 * HIDF);
    k_colstats<<<nStatBlk, 128, 0, stream>>>(bufB, b2, sums, sumsq, N);
    k_bn_relu<<<nFeatBlk, T, 0, stream>>>(bufB, b2, g2, be2, sums, sumsq, bufA, N);

    // ---- pool + classifier ----
    k_zero<<<(64 * HIDF + 64 + T - 1) / T, T, 0, stream>>>(pooled, 64 * HIDF + 64);
    k_pool<<<nFeatBlk, T, 0, stream>>>(bufA, bat, pooled, counts, N);
    k_classifier<<<1, 128, 0, stream>>>(pooled, counts, Wc1, bc1, Wc2, bc2, out);
}